// CayleyLearnedQuantizer_10170482557585
// MI455X (gfx1250) — compile-verified
//
#include <hip/hip_runtime.h>

typedef __attribute__((ext_vector_type(2))) float v2f;
typedef __attribute__((ext_vector_type(8))) float v8f;

#define DIM 128
#define LDSW 132              // padded LDS row stride (floats): rotates banks, 16B-aligned
#define ROWS_PER_BLOCK 128
#define AUGW 260              // 256 cols + 4 pad

// ---------------------------------------------------------------------------
// Kernel 1: Cayley map R = (I+A)^{-1} (I-A) via Gauss-Jordan on [I+A | I-A].
// Single workgroup, 1024 threads, barrier-synchronized elimination.
// ---------------------------------------------------------------------------
extern "C" __global__ __launch_bounds__(1024)
void cayley_build_r(const float* __restrict__ skew, float* __restrict__ Rout)
{
    extern __shared__ float smem[];
    float* Aug = smem;                 // 128 x 260
    float* fac = smem + DIM * AUGW;    // 128 elimination factors

    const int tid = threadIdx.x;

    // Init augmented matrix: left = I + A, right = I - A.
    // Strict-upper-tri param p for (i<j): p = i*(2D-i-1)/2 + (j-i-1)
    for (int it = 0; it < 32; ++it) {
        int idx = tid + 1024 * it;         // 0 .. 32767
        int i  = idx >> 8;                 // row
        int j  = idx & 255;                // augmented col
        int jj = j & 127;
        bool right = (j >= 128);
        float a = 0.0f;
        if (i != jj) {
            int lo = (i < jj) ? i : jj;
            int hi = (i < jj) ? jj : i;
            int p  = lo * (2 * DIM - lo - 1) / 2 + (hi - lo - 1);
            float s = skew[p];
            a = (i < jj) ? s : -s;
        }
        float val = ((i == jj) ? 1.0f : 0.0f) + (right ? -a : a);
        Aug[i * AUGW + j] = val;
    }
    __syncthreads();

    // Gauss-Jordan without explicit row scaling (divide at extraction).
    for (int k = 0; k < DIM; ++k) {
        if (tid < DIM) {
            float piv = Aug[k * AUGW + k];
            fac[tid] = (tid == k) ? 0.0f : Aug[tid * AUGW + k] / piv;
        }
        __syncthreads();
        {
            int i  = tid >> 3;             // 128 rows x 8 threads/row
            int c0 = (tid & 7) * 32;
            float f = fac[i];
            if (f != 0.0f) {
                float* rowi = Aug + i * AUGW;
                const float* rowk = Aug + k * AUGW;
                #pragma unroll 8
                for (int j = c0; j < c0 + 32; ++j)
                    rowi[j] -= f * rowk[j];
            }
        }
        __syncthreads();
    }

    // Extract R = right-half / diagonal.
    for (int it = 0; it < 16; ++it) {
        int idx = tid + 1024 * it;         // 0..16383
        int i = idx >> 7;
        int j = idx & 127;
        Rout[i * DIM + j] = Aug[i * AUGW + DIM + j] / Aug[i * AUGW + i];
    }
}

// ---------------------------------------------------------------------------
// Kernel 2: per-128-row tile: center, normalize, x@R^T, quantize, q@R,
// rescale, recenter. fp32 WMMA 16x16x4; 8 waves own disjoint 16-row M-tiles.
// ---------------------------------------------------------------------------
extern "C" __global__ __launch_bounds__(256)
void cayley_quant_main(const float* __restrict__ x,
                       const float* __restrict__ R,
                       const float* __restrict__ centroids,
                       const float* __restrict__ mean,
                       float* __restrict__ out)
{
    extern __shared__ float smem[];
    float* xq    = smem;                        // 128 x 132 (x tile, later Q)
    float* rr    = smem + DIM * LDSW;           // 128 x 132 (R, row-major)
    float* smean = rr + DIM * LDSW;             // 128
    float* snorm = smean + DIM;                 // 128
    float* scent = snorm + DIM;                 // 8

    const int tid  = threadIdx.x;
    const int lane = tid & 31;
    const int wave = tid >> 5;                  // 0..7
    const long long rowBase = (long long)blockIdx.x * ROWS_PER_BLOCK;

    // ---- Phase 0: stage x tile + R (float4 / ds_store_b128), mean, codebook
    {
        const float4* xg = (const float4*)(x + rowBase * DIM);
        const float4* rg = (const float4*)R;
        #pragma unroll
        for (int i = 0; i < 16; ++i) {
            int idx = tid + 256 * i;            // float4 index, 0..4095
            int row = idx >> 5;                 // 32 float4 per row
            int c4  = idx & 31;
            float4 v = xg[row * 32 + c4];
            *(float4*)(xq + row * LDSW + c4 * 4) = v;
            float4 w = rg[row * 32 + c4];
            *(float4*)(rr + row * LDSW + c4 * 4) = w;
        }
        if (tid < DIM) smean[tid] = mean[tid];
        if (tid < 8)   scent[tid] = centroids[tid];
    }
    __syncthreads();

    // ---- Phase 1: center + L2-normalize each row (threads 0..127)
    if (tid < ROWS_PER_BLOCK) {
        float* row = xq + tid * LDSW;
        float s = 0.0f;
        for (int k = 0; k < DIM; ++k) {
            float v = row[k] - smean[k];
            s += v * v;
        }
        float nrm = fmaxf(sqrtf(s), 1e-8f);
        snorm[tid] = nrm;
        float inv = 1.0f / nrm;
        for (int k = 0; k < DIM; ++k)
            row[k] = (row[k] - smean[k]) * inv;
    }
    __syncthreads();

    // Codebook + decision thresholds (sorted centroids => nearest == count of v>t_j)
    float c[8], th[7];
    #pragma unroll
    for (int i = 0; i < 8; ++i) c[i] = scent[i];
    #pragma unroll
    for (int i = 0; i < 7; ++i) th[i] = 0.5f * (c[i] + c[i + 1]);

    // WMMA fragment lane mapping (fp32 16x16x4, ISA 7.12.2):
    //   A: lane 0-15 -> M=lane, K={0,1}; lane 16-31 -> M=lane-16, K={2,3}
    //   B: mirrored over N; C/D: VGPR v -> M = v + 8*(lane>=16), N = lane&15
    const int m0    = wave * 16;
    const int am    = lane & 15;
    const int khalf = (lane >> 4) * 2;
    const int mhi   = (lane >> 4) << 3;

    // ---- GEMM1: x_rot = x_unit @ R^T  (B(k,n) = R[n][k], consecutive k -> b64)
    v8f q[8];
    {
        const float* aRow = xq + (m0 + am) * LDSW;
        #pragma unroll
        for (int nt = 0; nt < 8; ++nt) {
            const float* bRow = rr + (nt * 16 + am) * LDSW;   // R row indexed by k
            v8f acc = {};
            #pragma unroll 8
            for (int k = 0; k < DIM; k += 4) {
                v2f a = *(const v2f*)(aRow + k + khalf);
                v2f b = *(const v2f*)(bRow + k + khalf);
                acc = __builtin_amdgcn_wmma_f32_16x16x4_f32(
                          false, a, false, b, (short)0, acc, false, false);
            }
            // Quantize: forward value is exactly centroids[argmin |v - c|]
            v8f qq;
            #pragma unroll
            for (int e = 0; e < 8; ++e) {
                float v  = acc[e];
                float cv = c[0];
                #pragma unroll
                for (int j = 0; j < 7; ++j)
                    cv = (v > th[j]) ? c[j + 1] : cv;
                qq[e] = cv;
            }
            q[nt] = qq;
        }
    }

    // Overwrite own 16 rows of xq with Q (rows disjoint per wave: no barrier)
    #pragma unroll
    for (int nt = 0; nt < 8; ++nt) {
        #pragma unroll
        for (int e = 0; e < 8; ++e) {
            int ml = e + mhi;
            xq[(m0 + ml) * LDSW + nt * 16 + am] = q[nt][e];
        }
    }

    // ---- GEMM2: x_recon = (Q @ R) * norm + mean   (B(k,n) = R[k][n])
    {
        const float* aRow = xq + (m0 + am) * LDSW;
        #pragma unroll
        for (int nt = 0; nt < 8; ++nt) {
            const int n0 = nt * 16;
            v8f acc = {};
            #pragma unroll 8
            for (int k = 0; k < DIM; k += 4) {
                v2f a = *(const v2f*)(aRow + k + khalf);
                v2f b;
                b.x = rr[(k + khalf)     * LDSW + n0 + am];
                b.y = rr[(k + khalf + 1) * LDSW + n0 + am];
                acc = __builtin_amdgcn_wmma_f32_16x16x4_f32(
                          false, a, false, b, (short)0, acc, false, false);
            }
            #pragma unroll
            for (int e = 0; e < 8; ++e) {
                int ml   = e + mhi;
                int row  = m0 + ml;
                int n    = n0 + am;
                out[(rowBase + row) * DIM + n] = acc[e] * snorm[row] + smean[n];
            }
        }
    }
}

// ---------------------------------------------------------------------------
extern "C" void kernel_launch(void* const* d_in, const int* in_sizes, int n_in,
                              void* d_out, int out_size, void* d_ws, size_t ws_size,
                              hipStream_t stream)
{
    const float* x     = (const float*)d_in[0];   // (B, 128)
    const float* skew  = (const float*)d_in[1];   // 8128
    const float* cent  = (const float*)d_in[2];   // 8
    const float* rmean = (const float*)d_in[3];   // 128
    float* out = (float*)d_out;
    float* R   = (float*)d_ws;                    // 128*128 floats scratch

    (void)n_in; (void)out_size; (void)ws_size;

    // Kernel 1: build rotation matrix (one workgroup, ~130 KB LDS)
    size_t lds1 = (size_t)(DIM * AUGW + DIM) * sizeof(float);
    hipLaunchKernelGGL(cayley_build_r, dim3(1), dim3(1024), lds1, stream, skew, R);

    // Kernel 2: main transform, 128 rows per block (~136 KB LDS)
    int nrows  = in_sizes[0] / DIM;
    int blocks = nrows / ROWS_PER_BLOCK;
    size_t lds2 = (size_t)(2 * DIM * LDSW + 2 * DIM + 8) * sizeof(float);
    hipLaunchKernelGGL(cayley_quant_main, dim3(blocks), dim3(256), lds2, stream,
                       x, R, cent, rmean, out);
}